// MultiHeadAttention_22574348108615
// MI455X (gfx1250) — compile-verified
//
#include <hip/hip_runtime.h>
#include <math.h>
#include <cstdint>

// Problem constants (match reference)
#define BB 2
#define SS 2048
#define EE 768
#define HH 12
#define HD 64
#define E3 (3 * EE)          // 2304
#define HALF_WIN 256
#define NEGF (-9.0e15f)

// GEMM tiling
#define KC 32                // K-chunk staged in LDS
#define ASTR 36              // LDS row stride (floats): 36%64 banks -> conflict-free, 144B rows -> 16B aligned
#define SSTR 546             // attention score LDS stride (conflict-free, 8B aligned)

typedef float v2f __attribute__((ext_vector_type(2)));
typedef float v8f __attribute__((ext_vector_type(8)));

// fp32 WMMA: D(16x16) = A(16x4) * B(4x16) + C  — exact fp32, RNE, matches reference numerics
__device__ __forceinline__ v8f wmma_f32(v2f a, v2f b, v8f c) {
    return __builtin_amdgcn_wmma_f32_16x16x4_f32(false, a, false, b, (short)0, c, false, false);
}

// CDNA5 async global->LDS copy (ASYNCcnt path), with __has_builtin guard so the
// file compiles on either toolchain; fallback is a register double-buffer.
#if defined(__AMDGCN__) && __has_builtin(__builtin_amdgcn_global_load_async_to_lds_b128)
#define HAS_ASYNC_LDS 1
#else
#define HAS_ASYNC_LDS 0
#endif

#if HAS_ASYNC_LDS
// Parameter types per hipcc diagnostic: vector_size(16) int, AS1 (__device__) src,
// AS3 (__shared__) dst, imm offset, imm cpol.
typedef int v4i __attribute__((vector_size(4 * sizeof(int))));
typedef __attribute__((address_space(1))) v4i gv4i;
typedef __attribute__((address_space(3))) v4i lv4i;

__device__ __forceinline__ void async_copy_b128(const float* g, float* l) {
    __builtin_amdgcn_global_load_async_to_lds_b128(
        (gv4i*)(uintptr_t)g,   // generic global pointer bits == AS1 bits
        (lv4i*)l,              // addrspacecast generic -> LDS
        0, 0);
}
__device__ __forceinline__ void async_wait0() {
#if __has_builtin(__builtin_amdgcn_s_wait_asynccnt)
    __builtin_amdgcn_s_wait_asynccnt(0);
#else
    asm volatile("s_wait_asynccnt 0x0" ::: "memory");
#endif
}
#endif

// ---------------------------------------------------------------------------
// C[M,N] = A[M,K] @ W[N,K]^T + bias[N]
// Block (256 thr, 8 waves) -> 128x64 C tile. K staged in LDS in KC=32 chunks,
// double buffered: prefetch chunk k+1 (async-to-LDS when available) while the
// 8 waves run 32 WMMAs each on chunk k out of LDS.
// ---------------------------------------------------------------------------
__global__ __launch_bounds__(256)
void gemm_xwT(const float* __restrict__ A, const float* __restrict__ W,
              const float* __restrict__ bias, float* __restrict__ C,
              int M, int N, int K) {
    __shared__ float sA[2][128 * ASTR];   // 2*18432B
    __shared__ float sW[2][64 * ASTR];    // 2* 9216B  -> 55296B total

    const int tid  = threadIdx.x;
    const int lane = tid & 31;
    const int wv   = tid >> 5;            // 0..7
    const int half = lane >> 4;
    const int lm   = lane & 15;

    const int nt64 = N >> 6;
    const int mb   = blockIdx.x / nt64;
    const int nb   = blockIdx.x % nt64;
    const int m0   = mb << 7;             // 128-row block
    const int n0   = nb << 6;             // 64-col block

    // cooperative load mapping: 32 rows x 8 float4-cols per sweep
    const int lrow = tid >> 3;            // 0..31
    const int lcol = (tid & 7) << 2;      // 0,4,...,28 (float index)

    v8f acc[4] = {};

    auto compute = [&](int buf) {
#pragma unroll
        for (int kk = 0; kk < KC / 4; ++kk) {
            v2f a = *(const v2f*)&sA[buf][(wv * 16 + lm) * ASTR + 4 * kk + 2 * half];
#pragma unroll
            for (int j = 0; j < 4; ++j) {
                v2f b = *(const v2f*)&sW[buf][(j * 16 + lm) * ASTR + 4 * kk + 2 * half];
                acc[j] = wmma_f32(a, b, acc[j]);
            }
        }
    };

    const int nchunks = K / KC;

#if HAS_ASYNC_LDS
    auto prefetch = [&](int buf, int k0) {
#pragma unroll
        for (int i = 0; i < 4; ++i)       // A: 128x32
            async_copy_b128(A + (size_t)(m0 + lrow + 32 * i) * K + k0 + lcol,
                            &sA[buf][(lrow + 32 * i) * ASTR + lcol]);
#pragma unroll
        for (int i = 0; i < 2; ++i)       // W: 64x32
            async_copy_b128(W + (size_t)(n0 + lrow + 32 * i) * K + k0 + lcol,
                            &sW[buf][(lrow + 32 * i) * ASTR + lcol]);
    };

    prefetch(0, 0);
    async_wait0();
    __syncthreads();
    for (int kc = 0; kc < nchunks; ++kc) {
        const int cur = kc & 1;
        if (kc + 1 < nchunks) prefetch(cur ^ 1, (kc + 1) * KC);
        compute(cur);
        async_wait0();                    // my async writes to nxt landed in LDS
        __syncthreads();                  // everyone done reading cur / writing nxt
    }
#else
    float4 ra[4];
    float4 rw[2];
    auto loadregs = [&](int k0) {
#pragma unroll
        for (int i = 0; i < 4; ++i)
            ra[i] = *(const float4*)(A + (size_t)(m0 + lrow + 32 * i) * K + k0 + lcol);
#pragma unroll
        for (int i = 0; i < 2; ++i)
            rw[i] = *(const float4*)(W + (size_t)(n0 + lrow + 32 * i) * K + k0 + lcol);
    };
    auto storelds = [&](int buf) {
#pragma unroll
        for (int i = 0; i < 4; ++i)
            *(float4*)&sA[buf][(lrow + 32 * i) * ASTR + lcol] = ra[i];
#pragma unroll
        for (int i = 0; i < 2; ++i)
            *(float4*)&sW[buf][(lrow + 32 * i) * ASTR + lcol] = rw[i];
    };

    loadregs(0);
    storelds(0);
    __syncthreads();
    for (int kc = 0; kc < nchunks; ++kc) {
        const int cur  = kc & 1;
        const bool more = (kc + 1) < nchunks;
        if (more) loadregs((kc + 1) * KC);    // global->regs overlaps compute
        compute(cur);
        if (more) {
            __syncthreads();                  // all waves done reading buffers
            storelds(cur ^ 1);
            __syncthreads();                  // nxt visible to all
        }
    }
#endif

    // epilogue: bias + store (C/D layout: VGPR g -> row g (+8 for upper half-wave))
#pragma unroll
    for (int j = 0; j < 4; ++j) {
        const int col  = n0 + (j << 4) + lm;
        const float bc = bias[col];
#pragma unroll
        for (int g = 0; g < 8; ++g) {
            const int row = m0 + (wv << 4) + g + 8 * half;
            C[(size_t)row * N + col] = acc[j][g] + bc;
        }
    }
}

// ---------------------------------------------------------------------------
// Banded attention. One wave per (b, h, 16-query tile).
// qkv row layout (stride 3E): head h -> Q at h*192, K at +64, V at +128.
// Band limits keys to [q0-256, q0+271] -> <=33 key tiles (528 cols).
// Fragments are batched into registers before each WMMA burst so the compiler
// can issue one loadcnt wait per tile instead of one per WMMA.
// ---------------------------------------------------------------------------
__global__ __launch_bounds__(32)
void attn_banded(const float* __restrict__ qkv, const int* __restrict__ pmask,
                 float* __restrict__ vals) {
    __shared__ float s_p[16 * SSTR];
    __shared__ float red[32];
    __shared__ float denom[16];

    const int blk = blockIdx.x;
    const int qt  = blk & 127;            // S/16 = 128
    const int h   = (blk >> 7) % HH;
    const int b   = blk / (128 * HH);
    const int q0  = qt << 4;

    const int lane = threadIdx.x;
    const int half = lane >> 4;
    const int lm   = lane & 15;

    int k_lo = q0 - HALF_WIN; if (k_lo < 0) k_lo = 0;
    int k_hi = q0 + 16 + HALF_WIN; if (k_hi > SS) k_hi = SS;
    const int ncols  = k_hi - k_lo;       // multiple of 16, <= 528
    const int ntiles = ncols >> 4;

    const float* __restrict__ Qb = qkv + (size_t)(b * SS) * E3 + h * (3 * HD);
    const float* __restrict__ Kb = Qb + HD;
    const float* __restrict__ Vb = Qb + 2 * HD;

    // Preload Q A-fragments (row q0+lm, d = 4*kk + 2*half)
    v2f qf[16];
    {
        const float* qrow = Qb + (size_t)(q0 + lm) * E3 + 2 * half;
#pragma unroll
        for (int kk = 0; kk < 16; ++kk) qf[kk] = *(const v2f*)(qrow + 4 * kk);
    }

    int pmq[8];
#pragma unroll
    for (int g = 0; g < 8; ++g) pmq[g] = pmask[b * SS + q0 + g + 8 * half];

    const float scale = 0.125f;           // 1/sqrt(64)

    // ---- Pass A: banded score tiles -> LDS ----
    for (int t = 0; t < ntiles; ++t) {
        const int kbase = k_lo + (t << 4);
        const float* krow = Kb + (size_t)(kbase + lm) * E3 + 2 * half;
        if (t + 1 < ntiles) __builtin_prefetch(krow + (size_t)16 * E3, 0, 1);

        v2f kf[16];
#pragma unroll
        for (int kk = 0; kk < 16; ++kk) kf[kk] = *(const v2f*)(krow + 4 * kk);
        v8f acc = {};
#pragma unroll
        for (int kk = 0; kk < 16; ++kk) acc = wmma_f32(qf[kk], kf[kk], acc);

        const int kj  = kbase + lm;
        const int pmk = pmask[b * SS + kj];
#pragma unroll
        for (int g = 0; g < 8; ++g) {
            const int qi = q0 + g + 8 * half;
            const int d  = qi - kj;
            const bool ok = (d <= HALF_WIN) && (d >= -HALF_WIN) && (pmq[g] != 0) && (pmk != 0);
            s_p[(g + 8 * half) * SSTR + (t << 4) + lm] = ok ? acc[g] * scale : NEGF;
        }
    }
    __syncthreads();

    // ---- Softmax stats: row r = lm, each half-wave lane scans half the columns ----
    {
        const int hc = ncols >> 1;
        const int c0 = half * hc;
        float* row = &s_p[lm * SSTR];
        float mx = NEGF;
        for (int c = 0; c < hc; ++c) mx = fmaxf(mx, row[c0 + c]);
        red[lane] = mx;
        __syncthreads();
        mx = fmaxf(red[lm], red[lm + 16]);
        __syncthreads();
        float sum = 0.f;
        for (int c = 0; c < hc; ++c) {
            float e = __expf(row[c0 + c] - mx);
            row[c0 + c] = e;              // unnormalized probs back to LDS
            sum += e;
        }
        red[lane] = sum;
        __syncthreads();
        if (lane < 16) denom[lane] = red[lane] + red[lane + 16];
        __syncthreads();
    }

    // ---- Pass B: O = P @ V (A-frags from LDS, all V B-frags batched first) ----
    v8f oacc[4] = {};
    for (int t = 0; t < ntiles; ++t) {
        const int kbase = k_lo + (t << 4);

        v2f pa[4];
        v2f vf[4][4];
#pragma unroll
        for (int kk = 0; kk < 4; ++kk) {
            const int kofs = 4 * kk + 2 * half;
            pa[kk] = *(const v2f*)(&s_p[lm * SSTR + (t << 4) + kofs]);
            const float* v0 = Vb + (size_t)(kbase + kofs) * E3;
#pragma unroll
            for (int nc = 0; nc < 4; ++nc) {
                vf[kk][nc].x = v0[(nc << 4) + lm];
                vf[kk][nc].y = v0[E3 + (nc << 4) + lm];
            }
        }
#pragma unroll
        for (int kk = 0; kk < 4; ++kk)
#pragma unroll
            for (int nc = 0; nc < 4; ++nc)
                oacc[nc] = wmma_f32(pa[kk], vf[kk][nc], oacc[nc]);
    }

    // ---- Normalize rows and store vals [B*S, E] (col = h*64 + d) ----
#pragma unroll
    for (int g = 0; g < 8; ++g) {
        const int row = g + 8 * half;
        const float inv = 1.0f / denom[row];
        const size_t base = (size_t)(b * SS + q0 + row) * EE + h * HD;
#pragma unroll
        for (int nc = 0; nc < 4; ++nc)
            vals[base + (nc << 4) + lm] = oacc[nc][g] * inv;
    }
}

// ---------------------------------------------------------------------------
extern "C" void kernel_launch(void* const* d_in, const int* in_sizes, int n_in,
                              void* d_out, int out_size, void* d_ws, size_t ws_size,
                              hipStream_t stream) {
    const float* x     = (const float*)d_in[0];
    const int*   pmask = (const int*)  d_in[1];
    const float* Wqkv  = (const float*)d_in[2];
    const float* bqkv  = (const float*)d_in[3];
    const float* Wo    = (const float*)d_in[4];
    const float* bo    = (const float*)d_in[5];
    float* out = (float*)d_out;

    float* qkv  = (float*)d_ws;                   // [B*S, 3E]  37.75 MB
    float* vals = qkv + (size_t)BB * SS * E3;     // [B*S, E]   12.6 MB

    const int M = BB * SS;                        // 4096

    // QKV projection: [4096,2304] = x @ Wqkv^T + bqkv
    gemm_xwT<<<(M >> 7) * (E3 >> 6), 256, 0, stream>>>(x, Wqkv, bqkv, qkv, M, E3, EE);

    // Banded attention
    attn_banded<<<BB * HH * (SS >> 4), 32, 0, stream>>>(qkv, pmask, vals);

    // Output projection: [4096,768] = vals @ Wo^T + bo
    gemm_xwT<<<(M >> 7) * (EE >> 6), 256, 0, stream>>>(vals, Wo, bo, out, M, EE, EE);
}